// ReconBlock_90563680403822
// MI455X (gfx1250) — compile-verified
//
#include <hip/hip_runtime.h>

// 4x pixel-shuffle (depth-to-space):
//   out[b][co][h*4+rh][w*4+rw] = in[b][co*16 + rh*4 + rw][h][w]
// B=32, C=400, c_out=25, H=W=64 -> 256x256. Pure data movement, HBM-bound
// (~420 MB total -> ~18 us floor at 23.3 TB/s). Strategy: stage 16 channels
// x 64 floats per (b,co,h) slice into LDS with the gfx1250 async-to-LDS
// path (ASYNCcnt), transpose out of LDS (bank-conflict-free), store
// contiguous non-temporal b128.

typedef float v4f __attribute__((ext_vector_type(4)));
typedef int   v4i __attribute__((ext_vector_type(4)));

#define AS1 __attribute__((address_space(1)))
#define AS3 __attribute__((address_space(3)))

#if defined(__AMDGCN__) && __has_builtin(__builtin_amdgcn_global_load_async_to_lds_b128)
#define USE_ASYNC_LDS 1
#else
#define USE_ASYNC_LDS 0
#endif

__global__ __launch_bounds__(256) void pixel_shuffle4_kernel(
    const float* __restrict__ in, float* __restrict__ out) {
  __shared__ float tile[16 * 64];  // 16 channels x 64 w-values = 4 KB

  const int t   = threadIdx.x;     // 0..255
  const int blk = blockIdx.x;      // (b, co, h) linearized
  const int h   = blk & 63;        // 0..63
  const int bc  = blk >> 6;
  const int co  = bc % 25;         // 0..24
  const int b   = bc / 25;         // 0..31

  // ---- stage: 16 input channel-rows -> LDS (16 B per lane, coalesced) ----
  const int ch = t >> 4;           // 0..15  channel within the group of 16
  const int qw = (t & 15) << 2;    // 0,4,...,60 starting w
  const size_t src =
      ((((size_t)b * 400) + (size_t)co * 16 + (size_t)ch) * 64 + (size_t)h) * 64 + (size_t)qw;

#if USE_ASYNC_LDS
  // gfx1250 async copy: memory -> LDS without a VGPR round-trip (ASYNCcnt).
  __builtin_amdgcn_global_load_async_to_lds_b128(
      (AS1 v4i*)(in + src), (AS3 v4i*)&tile[t * 4], /*offset=*/0, /*cpol=*/0);
#if __has_builtin(__builtin_amdgcn_s_wait_asynccnt)
  __builtin_amdgcn_s_wait_asynccnt(0);
#else
  asm volatile("s_wait_asynccnt 0" ::: "memory");
#endif
#else
  const v4f v = __builtin_nontemporal_load((const v4f*)(in + src));
  *(v4f*)&tile[t * 4] = v;
#endif
  __syncthreads();

  // ---- permute out of LDS; each read hits 32 consecutive words (distinct banks) per wave --------
  const int rh = t >> 6;           // 0..3
  const int w  = t & 63;           // 0..63
  v4f o;
  o.x = tile[(rh * 4 + 0) * 64 + w];
  o.y = tile[(rh * 4 + 1) * 64 + w];
  o.z = tile[(rh * 4 + 2) * 64 + w];
  o.w = tile[(rh * 4 + 3) * 64 + w];

  // out flat index: ((b*25 + co)*256 + h*4+rh)*256 + w*4  (16 B/lane, waves
  // write fully contiguous 512 B runs)
  const size_t dst =
      ((((size_t)b * 25) + (size_t)co) * 256 + (size_t)(h * 4 + rh)) * 256 + (size_t)w * 4;
  __builtin_nontemporal_store(o, (v4f*)(out + dst));
}

extern "C" void kernel_launch(void* const* d_in, const int* in_sizes, int n_in,
                              void* d_out, int out_size, void* d_ws, size_t ws_size,
                              hipStream_t stream) {
  const float* in = (const float*)d_in[0];
  float* out      = (float*)d_out;
  // one block per (b, co, h): 32 * 25 * 64 = 51200 blocks of 256 threads
  const int blocks = 32 * 25 * 64;
  pixel_shuffle4_kernel<<<blocks, 256, 0, stream>>>(in, out);
}